// TimeAwarePointHead_16836271800624
// MI455X (gfx1250) — compile-verified
//
#include <hip/hip_runtime.h>
#include <hip/hip_bf16.h>

typedef __attribute__((ext_vector_type(2))) float v2f;
typedef __attribute__((ext_vector_type(8))) float v8f;

#define B_    4
#define C_    16
#define D_    32
#define H_    256
#define W_    256
#define N_    500000
#define P_    16
#define TH_   8
#define HID_  32
#define INDIM_ 40
#define KSTR  44                    // LDS row stride (pad 40->44: conflict-free b64 across 16 rows)
#define WPB   8                     // waves per block
#define TILES (N_ / 16)             // 31250, exact
#define NBLK  ((TILES + WPB - 1) / WPB)
#define NKC   10                    // K chunks of 4 (10*4 = 40)

// workspace layout (floats)
#define WS_TF    0
#define WS_SUM   8
#define WS_SQ    (WS_SUM + B_ * HID_)
#define WS_SCALE (WS_SQ + B_ * HID_)
#define WS_SHIFT (WS_SCALE + B_ * HID_)

__global__ void k_init(const float* __restrict__ w_time, const float* __restrict__ b_time,
                       const int* __restrict__ time_idx, float* __restrict__ ws) {
  int t = threadIdx.x;
  if (t < 2 * B_ * HID_) ws[WS_SUM + t] = 0.0f;           // zero sum + sumsq accumulators
  if (t < TH_) {
    float v = w_time[time_idx[0] * TH_ + t] + b_time[t];  // one-hot @ W_time + b
    ws[WS_TF + t] = v > 0.0f ? v : 0.0f;                  // relu
  }
}

__global__ void k_stats(const float* __restrict__ gamma, const float* __restrict__ beta,
                        float* __restrict__ ws) {
  int t = threadIdx.x;                                    // t = b*32 + n, 128 threads
  if (t < B_ * HID_) {
    const float inv_n = 1.0f / (float)N_;
    float mean = ws[WS_SUM + t] * inv_n;
    float var  = ws[WS_SQ + t] * inv_n - mean * mean;     // biased variance (torch BN)
    float sc   = gamma[t & (HID_ - 1)] * rsqrtf(var + 1e-5f);
    ws[WS_SCALE + t] = sc;
    ws[WS_SHIFT + t] = beta[t & (HID_ - 1)] - mean * sc;
  }
}

template <int PASS>
__global__ __launch_bounds__(256) void mlp_pass(
    const float* __restrict__ vf, const int* __restrict__ coords,
    const float* __restrict__ pf, const float* __restrict__ w1,
    const float* __restrict__ bias1, const float* __restrict__ w2,
    const float* __restrict__ b2, float* __restrict__ ws,
    float* __restrict__ out) {
  __shared__ float sX[WPB][16 * KSTR];       // per-wave 16x40 x-tile (row-major, padded)
  __shared__ float sW1p[NKC * 2 * 32 * 2];   // W1 pre-swizzled as WMMA B fragments:
                                             // [kc][ntile][lane] -> float2 (kr, kr+1)
  __shared__ float sSum[2 * HID_];           // pass1: [sum(32), sumsq(32)]
  __shared__ float sW2[HID_ * 3 + 4];        // pass2: w2 (32x3) + b2
  __shared__ float sSS[2 * HID_];            // pass2: [scale(32), shift(32)] for batch b

  const int b     = blockIdx.y;
  const int tid   = threadIdx.x;
  const int wave  = tid >> 5;
  const int lane  = tid & 31;
  const int half  = lane >> 4;               // 0: K pair {4k,4k+1} / 1: {4k+2,4k+3}
  const int col16 = lane & 15;

  // ---- swizzle W1 into B-fragment order: one b64 per lane per (kc, ntile) ----
  for (int e = tid; e < NKC * 2 * 32; e += 256) {   // 640 float2 entries
    const int kc = e >> 6;                          // e / 64
    const int r  = e & 63;
    const int t  = r >> 5;                          // n-tile
    const int le = r & 31;                          // lane this entry serves
    const int kr = kc * 4 + (le >> 4) * 2;
    const int cl = t * 16 + (le & 15);
    sW1p[2 * e + 0] = w1[kr * HID_ + cl];
    sW1p[2 * e + 1] = w1[(kr + 1) * HID_ + cl];
  }
  if (PASS == 1) {
    if (tid < 2 * HID_) sSum[tid] = 0.0f;
  }
  if (PASS == 2) {
    if (tid < HID_ * 3) sW2[tid] = w2[tid];
    if (tid < 4) sW2[HID_ * 3 + tid] = (tid < 3) ? b2[tid] : 0.0f;
    if (tid < 2 * HID_) {
      int nn = tid & (HID_ - 1);
      sSS[tid] = (tid < HID_) ? ws[WS_SCALE + b * HID_ + nn]
                              : ws[WS_SHIFT + b * HID_ + nn];
    }
  }

  const int  tileRaw = blockIdx.x * WPB + wave;
  const bool valid   = (tileRaw < TILES);
  const int  tile    = valid ? tileRaw : (TILES - 1);  // clamp: keep EXEC full at WMMA
  const int  pt      = tile * 16 + col16;

  // ---- stage x[16][40] in LDS: [voxel gather(16) | point feats(16) | time(8)] ----
  float* xrow = &sX[wave][col16 * KSTR];
  {
    const int ci = (b * N_ + pt) * 3;
    const int cx = coords[ci + 0];
    const int cy = coords[ci + 1];
    const int cz = coords[ci + 2];
    const float* vfb = vf + (size_t)b * C_ * (D_ * H_ * W_) +
                       ((size_t)cz * H_ + cy) * (size_t)W_ + cx;
#pragma unroll
    for (int c = 0; c < 8; ++c) {            // each lane gathers 8 channels of its point
      int ch = half * 8 + c;
      xrow[ch] = vfb[(size_t)ch * (D_ * H_ * W_)];
    }
    const float4* pfv = (const float4*)(pf + ((size_t)b * N_ + pt) * P_ + half * 8);
    float4 f0 = pfv[0], f1 = pfv[1];
    *(float4*)&xrow[C_ + half * 8 + 0] = f0;
    *(float4*)&xrow[C_ + half * 8 + 4] = f1;
    if (half == 0) {
#pragma unroll
      for (int k = 0; k < TH_; ++k) xrow[C_ + P_ + k] = ws[WS_TF + k];
    }
  }
  __syncthreads();

  // ---- GEMM1: h[16x32] = x[16x40] @ w1[40x32] + b1 via V_WMMA_F32_16X16X4_F32 ----
  const float bA = bias1[col16];
  const float bB = bias1[16 + col16];
  v8f acc0 = {bA, bA, bA, bA, bA, bA, bA, bA};   // C/D layout: lane owns column n=col16
  v8f acc1 = {bB, bB, bB, bB, bB, bB, bB, bB};
  const v2f* w1p = (const v2f*)sW1p;
#pragma unroll
  for (int kc = 0; kc < NKC; ++kc) {
    const int kr = kc * 4 + half * 2;
    v2f a   = *(const v2f*)&sX[wave][col16 * KSTR + kr];  // A frag: row M=col16, K pair
    v2f bb0 = w1p[kc * 64 + lane];                        // B frag n-tile 0 (b64 pair)
    v2f bb1 = w1p[kc * 64 + 32 + lane];                   // B frag n-tile 1 (offset1:32)
    acc0 = __builtin_amdgcn_wmma_f32_16x16x4_f32(false, a, false, bb0, (short)0, acc0,
                                                 false, false);
    acc1 = __builtin_amdgcn_wmma_f32_16x16x4_f32(false, a, false, bb1, (short)0, acc1,
                                                 false, false);
  }

  if (PASS == 1) {
    // BN statistics: per-lane partials over 8 rows, LDS-atomic reduce, one global atomic/ch
    if (valid) {
      float s0 = 0.f, q0 = 0.f, s1 = 0.f, q1 = 0.f;
#pragma unroll
      for (int r = 0; r < 8; ++r) {
        float h0 = acc0[r], h1 = acc1[r];
        s0 += h0; q0 += h0 * h0;
        s1 += h1; q1 += h1 * h1;
      }
      atomicAdd(&sSum[col16], s0);
      atomicAdd(&sSum[16 + col16], s1);
      atomicAdd(&sSum[HID_ + col16], q0);
      atomicAdd(&sSum[HID_ + 16 + col16], q1);
    }
    __syncthreads();
    if (tid < 2 * HID_) {
      float v = sSum[tid];
      float* dst = (tid < HID_) ? &ws[WS_SUM + b * HID_ + tid]
                                : &ws[WS_SQ + b * HID_ + (tid - HID_)];
      atomicAdd(dst, v);
    }
  }

  if (PASS == 2) {
    // BN affine + ReLU, transpose through LDS, then 32x3 tail GEMM (96 FMAs/point)
    const int n0 = col16, n1 = 16 + col16;
    const float sc0 = sSS[n0], sc1 = sSS[n1];
    const float sh0 = sSS[HID_ + n0], sh1 = sSS[HID_ + n1];
    float* sY = &sX[wave][0];   // reuse x-tile LDS as y[16][32]
#pragma unroll
    for (int r = 0; r < 8; ++r) {
      const int m = r + half * 8;
      float y0 = acc0[r] * sc0 + sh0; y0 = y0 > 0.f ? y0 : 0.f;
      float y1 = acc1[r] * sc1 + sh1; y1 = y1 > 0.f ? y1 : 0.f;
      sY[m * HID_ + n0] = y0;
      sY[m * HID_ + n1] = y1;
    }
    __syncthreads();
    if (valid && lane < 16) {
      const int m = lane;
      float f0 = sW2[HID_ * 3 + 0];
      float f1 = sW2[HID_ * 3 + 1];
      float f2 = sW2[HID_ * 3 + 2];
#pragma unroll
      for (int n = 0; n < HID_; ++n) {
        float y = sY[m * HID_ + n];
        f0 += y * sW2[n * 3 + 0];
        f1 += y * sW2[n * 3 + 1];
        f2 += y * sW2[n * 3 + 2];
      }
      const size_t o = ((size_t)b * N_ + (size_t)tile * 16 + m) * 3;
      out[o + 0] = f0;
      out[o + 1] = f1;
      out[o + 2] = f2;
    }
  }
}

extern "C" void kernel_launch(void* const* d_in, const int* in_sizes, int n_in,
                              void* d_out, int out_size, void* d_ws, size_t ws_size,
                              hipStream_t stream) {
  const float* vf      = (const float*)d_in[0];
  const int*   coords  = (const int*)d_in[1];
  const float* pf      = (const float*)d_in[2];
  const float* w_time  = (const float*)d_in[3];
  const float* b_time  = (const float*)d_in[4];
  const float* w1      = (const float*)d_in[5];
  const float* b1      = (const float*)d_in[6];
  const float* gamma   = (const float*)d_in[7];
  const float* beta    = (const float*)d_in[8];
  const float* w2      = (const float*)d_in[9];
  const float* b2      = (const float*)d_in[10];
  const int*   tidx    = (const int*)d_in[11];
  float* ws  = (float*)d_ws;
  float* out = (float*)d_out;

  k_init<<<1, 256, 0, stream>>>(w_time, b_time, tidx, ws);
  dim3 grid(NBLK, B_);
  mlp_pass<1><<<grid, 256, 0, stream>>>(vf, coords, pf, w1, b1, w2, b2, ws, out);
  k_stats<<<1, 128, 0, stream>>>(gamma, beta, ws);
  mlp_pass<2><<<grid, 256, 0, stream>>>(vf, coords, pf, w1, b1, w2, b2, ws, out);
}